// NormalizeSample_30167850287224
// MI455X (gfx1250) — compile-verified
//
#include <hip/hip_runtime.h>
#include <hip/hip_bf16.h>
#include <math.h>

typedef __attribute__((ext_vector_type(2))) float v2f;
typedef __attribute__((ext_vector_type(8))) float v8f;

// ---------------------------------------------------------------------------
// Kernel 1: per-block partial reduction of (sum, sum of squares, nonzero count)
// using V_WMMA_F32_16X16X4_F32 with an all-ones B matrix as the reduction
// engine. Each WMMA reduces 64 f32 values (dot-4 accumulate per C slot).
// All 16 columns of C are identical; summing every lane's 8 C values across
// the block yields 16x the true total, so we scale by 1/16 (exact in fp).
// ---------------------------------------------------------------------------
__global__ void __launch_bounds__(256)
fish_reduce_kernel(const float4* __restrict__ x,
                   float* __restrict__ psum,
                   float* __restrict__ psq,
                   float* __restrict__ pcnt,
                   int f4_per_block, int iters) {
    const int t = threadIdx.x;
    const long base = (long)blockIdx.x * (long)f4_per_block;

    v8f cs = {};   // sum accumulator
    v8f cq = {};   // sum-of-squares accumulator
    v8f cc = {};   // nonzero-count accumulator
    const v2f ones = {1.0f, 1.0f};

    for (int i = 0; i < iters; ++i) {
        float4 v = x[base + t + (long)i * 256];

        v2f a0 = {v.x, v.y};
        v2f a1 = {v.z, v.w};
        v2f q0 = a0 * a0;
        v2f q1 = a1 * a1;
        v2f m0 = {v.x != 0.0f ? 1.0f : 0.0f, v.y != 0.0f ? 1.0f : 0.0f};
        v2f m1 = {v.z != 0.0f ? 1.0f : 0.0f, v.w != 0.0f ? 1.0f : 0.0f};

        cs = __builtin_amdgcn_wmma_f32_16x16x4_f32(false, a0, false, ones, (short)0, cs, false, false);
        cs = __builtin_amdgcn_wmma_f32_16x16x4_f32(false, a1, false, ones, (short)0, cs, false, false);
        cq = __builtin_amdgcn_wmma_f32_16x16x4_f32(false, q0, false, ones, (short)0, cq, false, false);
        cq = __builtin_amdgcn_wmma_f32_16x16x4_f32(false, q1, false, ones, (short)0, cq, false, false);
        cc = __builtin_amdgcn_wmma_f32_16x16x4_f32(false, m0, false, ones, (short)0, cc, false, false);
        cc = __builtin_amdgcn_wmma_f32_16x16x4_f32(false, m1, false, ones, (short)0, cc, false, false);
    }

    float s = 0.0f, q = 0.0f, n = 0.0f;
#pragma unroll
    for (int j = 0; j < 8; ++j) { s += cs[j]; q += cq[j]; n += cc[j]; }

    __shared__ float ls[256];
    __shared__ float lq[256];
    __shared__ float lc[256];
    ls[t] = s; lq[t] = q; lc[t] = n;
    __syncthreads();
#pragma unroll
    for (int off = 128; off > 0; off >>= 1) {
        if (t < off) {
            ls[t] += ls[t + off];
            lq[t] += lq[t + off];
            lc[t] += lc[t + off];
        }
        __syncthreads();
    }
    if (t == 0) {
        psum[blockIdx.x] = ls[0] * (1.0f / 16.0f);
        psq [blockIdx.x] = lq[0] * (1.0f / 16.0f);
        pcnt[blockIdx.x] = lc[0] * (1.0f / 16.0f);
    }
}

// ---------------------------------------------------------------------------
// Kernel 2: per-sample final stats. 64 partials per sample -> mean, 1/std
// (unbiased, ddof=1). Done in double; negligible cost.
// ---------------------------------------------------------------------------
__global__ void __launch_bounds__(64)
fish_stats_kernel(const float* __restrict__ psum,
                  const float* __restrict__ psq,
                  const float* __restrict__ pcnt,
                  float* __restrict__ mean,
                  float* __restrict__ invstd,
                  int bps) {
    const int s = blockIdx.x;
    const int t = threadIdx.x;
    __shared__ float a[64];
    __shared__ float b[64];
    __shared__ float c[64];
    a[t] = psum[s * bps + t];
    b[t] = psq [s * bps + t];
    c[t] = pcnt[s * bps + t];
    __syncthreads();
#pragma unroll
    for (int off = 32; off > 0; off >>= 1) {
        if (t < off) {
            a[t] += a[t + off];
            b[t] += b[t + off];
            c[t] += c[t + off];
        }
        __syncthreads();
    }
    if (t == 0) {
        double S = (double)a[0];
        double Q = (double)b[0];
        double N = (double)c[0];
        double m = S / N;
        double var = (Q - S * S / N) / (N - 1.0);
        mean[s]   = (float)m;
        invstd[s] = (float)(1.0 / sqrt(var));
    }
}

// ---------------------------------------------------------------------------
// Kernel 3: elementwise normalize. Blocks map to contiguous chunks within one
// sample, so mean/invstd are block-uniform (scalar) loads. 8x float4 / thread.
// ---------------------------------------------------------------------------
__global__ void __launch_bounds__(256)
fish_norm_kernel(const float4* __restrict__ x,
                 float4* __restrict__ out,
                 const float* __restrict__ mean,
                 const float* __restrict__ invstd,
                 int blocks_per_sample, int f4_per_block) {
    const int s = blockIdx.x / blocks_per_sample;
    const float m   = mean[s];
    const float inv = invstd[s];
    const long base = (long)blockIdx.x * (long)f4_per_block;
    const int t = threadIdx.x;
#pragma unroll
    for (int i = 0; i < 8; ++i) {
        long idx = base + t + (long)i * 256;
        float4 v = x[idx];
        float4 r;
        r.x = (v.x != 0.0f) ? (v.x - m) * inv : v.x;
        r.y = (v.y != 0.0f) ? (v.y - m) * inv : v.y;
        r.z = (v.z != 0.0f) ? (v.z - m) * inv : v.z;
        r.w = (v.w != 0.0f) ? (v.w - m) * inv : v.w;
        out[idx] = r;
    }
}

extern "C" void kernel_launch(void* const* d_in, const int* in_sizes, int n_in,
                              void* d_out, int out_size, void* d_ws, size_t ws_size,
                              hipStream_t stream) {
    const float* x = (const float*)d_in[0];
    float* out = (float*)d_out;

    const long n  = (long)in_sizes[0];   // 64*3*512*512 = 50331648
    const int  NS = 64;                  // samples
    const long per   = n / NS;           // 786432 elements / sample
    const long f4ps  = per / 4;          // 196608 float4 / sample

    // Kernel 1 decomposition: 64 blocks/sample, 256 thr, 12 float4/thr
    const int  T      = 256;
    const int  BPS1   = 64;
    const int  f4pb1  = (int)(f4ps / BPS1);   // 3072
    const int  iters1 = f4pb1 / T;            // 12
    const int  nb1    = NS * BPS1;            // 4096 blocks

    // Workspace layout (floats): psum[nb1] | psq[nb1] | pcnt[nb1] | mean[NS] | invstd[NS]
    float* ws     = (float*)d_ws;
    float* psum   = ws;
    float* psq    = ws + nb1;
    float* pcnt   = ws + 2 * nb1;
    float* mean   = ws + 3 * nb1;
    float* invstd = mean + NS;

    fish_reduce_kernel<<<nb1, T, 0, stream>>>((const float4*)x, psum, psq, pcnt,
                                              f4pb1, iters1);

    fish_stats_kernel<<<NS, BPS1, 0, stream>>>(psum, psq, pcnt, mean, invstd, BPS1);

    // Kernel 3 decomposition: 2048 float4/block -> 96 blocks/sample
    const int F4PB3 = 2048;
    const int bps3  = (int)(f4ps / F4PB3);    // 96
    fish_norm_kernel<<<NS * bps3, T, 0, stream>>>((const float4*)x, (float4*)out,
                                                  mean, invstd, bps3, F4PB3);
}